// NegativeLogLikelihood_32598801776936
// MI455X (gfx1250) — compile-verified
//
#include <hip/hip_runtime.h>
#include <hip/hip_bf16.h>

typedef __attribute__((ext_vector_type(2))) float v2f;
typedef __attribute__((ext_vector_type(8))) float v8f;

#define NTOT 16384
#define BLK  1024
#define EPT  (NTOT / BLK)   // 16 elements per thread

// Single-workgroup Cox NLL: exp -> 3-level prefix scan -> masked log reduction.
// Cross-wave scan of the 32 wave sums is done with V_WMMA_F32_16X16X4_F32:
//   Y = L * X, L = 16x16 lower-triangular ones (split into four 16x4 K-chunks),
//   X column 0 = wave sums 0..15, column 1 = wave sums 16..31.
__global__ __launch_bounds__(BLK) void coxnll_kernel(const float* __restrict__ pred,
                                                     const float* __restrict__ label,
                                                     float* __restrict__ out) {
    __shared__ float s_wsum[32];     // per-wave totals
    __shared__ float s_colscan[32];  // Y[:,0] in [0..15], Y[:,1] in [16..31]
    __shared__ float s_woff[32];     // exclusive offsets per wave
    __shared__ float s_redA[32];
    __shared__ float s_redB[32];

    const int tid  = threadIdx.x;
    const int lane = tid & 31;
    const int wave = tid >> 5;
    const int base = tid * EPT;

    // ---- load 16 contiguous preds as 4x float4 (global_load_b128) ----
    float p[EPT];
    const float4* p4 = (const float4*)(pred + base);
#pragma unroll
    for (int v = 0; v < EPT / 4; ++v) {
        float4 q = p4[v];
        p[4 * v + 0] = q.x; p[4 * v + 1] = q.y;
        p[4 * v + 2] = q.z; p[4 * v + 3] = q.w;
    }

    // ---- level 0: per-thread inclusive scan of exp(pred) ----
    float ls[EPT];
    float run = 0.0f;
#pragma unroll
    for (int i = 0; i < EPT; ++i) { run += __expf(p[i]); ls[i] = run; }
    const float total = run;

    // ---- level 1: wave32 shuffle inclusive scan of per-thread totals ----
    float incl = total;
#pragma unroll
    for (int d = 1; d < 32; d <<= 1) {
        float up = __shfl_up(incl, d, 32);
        if (lane >= d) incl += up;
    }
    if (lane == 31) s_wsum[wave] = incl;
    __syncthreads();

    // ---- level 2: wave 0 scans the 32 wave sums via WMMA triangular matmul ----
    if (wave == 0) {
        const int   mrow  = lane & 15;              // M for A, N for B/C/D
        const bool  hi    = lane >= 16;
        const bool  liveN = (mrow < 2);             // only columns 0,1 carry data
        const int   colsel = 16 * (mrow & 1);       // uniform, in-range LDS base
        v8f acc = {};                               // C starts at zero
#pragma unroll
        for (int b = 0; b < 4; ++b) {
            // A chunk = L[:, 4b..4b+3]; A layout: v0 = {K0 lo-lanes | K2 hi-lanes},
            //                                     v1 = {K1 lo-lanes | K3 hi-lanes}
            const int k0 = 4 * b + (hi ? 2 : 0);
            const int k1 = 4 * b + (hi ? 3 : 1);
            v2f a, bm;
            a.x = (k0 <= mrow) ? 1.0f : 0.0f;
            a.y = (k1 <= mrow) ? 1.0f : 0.0f;
            // B chunk = X[4b..4b+3, :]; X[k][0]=wsum[k], X[k][1]=wsum[16+k], else 0.
            // Branch-free: unconditional ds_load from an always-valid address,
            // then v_cndmask the dead columns to zero (no exec-mask churn).
            const float b0 = s_wsum[colsel + k0];
            const float b1 = s_wsum[colsel + k1];
            bm.x = liveN ? b0 : 0.0f;
            bm.y = liveN ? b1 : 0.0f;
            acc = __builtin_amdgcn_wmma_f32_16x16x4_f32(
                /*neg_a=*/false, a, /*neg_b=*/false, bm,
                /*c_mod=*/(short)0, acc, /*reuse_a=*/false, /*reuse_b=*/false);
        }
        // C/D layout: VGPR r -> M=r (lanes 0-15), M=r+8 (lanes 16-31); N = lane%16.
        if (liveN) {
#pragma unroll
            for (int r = 0; r < 8; ++r) {
                const int M = r + (hi ? 8 : 0);
                s_colscan[mrow * 16 + M] = acc[r];
            }
        }
    }
    __syncthreads();

    // stitch column-1 carry and build exclusive per-wave offsets
    if (tid == 0) {
        const float carry = s_colscan[15];          // full sum of waves 0..15
        s_woff[0] = 0.0f;
        for (int j = 1; j < 32; ++j)
            s_woff[j] = (j <= 16) ? s_colscan[j - 1] : (s_colscan[j - 1] + carry);
    }
    __syncthreads();

    // ---- final pass: masked reduction of pred - log(prefix) ----
    const float excl = s_woff[wave] + (incl - total);   // exclusive thread offset
    const float2* lab2 = (const float2*)label;          // (ytime, yevent) pairs
    float sum_d = 0.0f, cnt = 0.0f;
#pragma unroll
    for (int i = 0; i < EPT; ++i) {
        const float ev  = lab2[base + i].y;             // yevent
        const float pre = excl + ls[i];                 // inclusive cumsum(exp)
        sum_d += ev * (p[i] - __logf(pre));
        cnt   += ev;
    }

    // wave reduce, then cross-wave reduce
#pragma unroll
    for (int d = 16; d >= 1; d >>= 1) {
        sum_d += __shfl_xor(sum_d, d, 32);
        cnt   += __shfl_xor(cnt,   d, 32);
    }
    if (lane == 0) { s_redA[wave] = sum_d; s_redB[wave] = cnt; }
    __syncthreads();

    if (tid == 0) {
        float S = 0.0f, C = 0.0f;
        for (int w = 0; w < 32; ++w) { S += s_redA[w]; C += s_redB[w]; }
        out[0] = (C == 0.0f) ? 0.0f : -(S / fmaxf(C, 1.0f));  // cost
        out[1] = C;                                           // n_observed
    }
}

extern "C" void kernel_launch(void* const* d_in, const int* in_sizes, int n_in,
                              void* d_out, int out_size, void* d_ws, size_t ws_size,
                              hipStream_t stream) {
    const float* pred  = (const float*)d_in[0];   // (16384, 1) f32
    const float* label = (const float*)d_in[1];   // (16384, 2) f32: [ytime, yevent]
    float* out = (float*)d_out;                   // [cost, n_observed]
    (void)in_sizes; (void)n_in; (void)out_size; (void)d_ws; (void)ws_size;
    coxnll_kernel<<<1, BLK, 0, stream>>>(pred, label, out);
}